// RNN_66297115181246
// MI455X (gfx1250) — compile-verified
//
#include <hip/hip_runtime.h>

#define H      64
#define G      256          // 4*H gates
#define CHUNK  128          // timesteps of x staged per LDS chunk (2*CHUNK == 256 floats)

typedef __attribute__((ext_vector_type(2))) float v2f;
typedef __attribute__((ext_vector_type(8))) float v8f;

__device__ __forceinline__ v2f mkv2(float a, float b) { v2f r; r.x = a; r.y = b; return r; }

// fast sigmoid/tanh: v_exp_f32 + v_rcp_f32 on the critical path
__device__ __forceinline__ float fsigmoid(float x) {
    return __builtin_amdgcn_rcpf(1.0f + __expf(-x));
}
__device__ __forceinline__ float ftanh(float x) {
    return 2.0f * __builtin_amdgcn_rcpf(1.0f + __expf(-2.0f * x)) - 1.0f;
}

// ---------------------------------------------------------------------------
// Kernel A: the sequential LSTM scan. One workgroup = 8 wave32 on one WGP.
// Thread j (0..255) owns gate j: W_hh row j in 64 VGPRs, dot with h (LDS,
// uniform-address broadcast reads, b128) using packed v_pk_fma_f32.
// Threads 0..63 then apply the nonlinearities and update c/h.
// ---------------------------------------------------------------------------
__global__ __launch_bounds__(256, 1)
void lstm_scan_kernel(const float* __restrict__ x,
                      const float* __restrict__ W_ih,
                      const float* __restrict__ W_hh,
                      const float* __restrict__ b_ih,
                      const float* __restrict__ b_hh,
                      float* __restrict__ hs,
                      int T) {
    __shared__ float h_lds[H];
    __shared__ float gates_lds[G];
    __shared__ float xbuf[2 * CHUNK];

    const int j = threadIdx.x;

    // Recurrent weight row in registers: 64 f32 = 32 packed v2f.
    v2f w2[H / 2];
    {
        const float4* wrow = (const float4*)(W_hh + j * H);
        #pragma unroll
        for (int k = 0; k < H / 4; ++k) {
            float4 w = wrow[k];
            w2[2 * k + 0] = mkv2(w.x, w.y);
            w2[2 * k + 1] = mkv2(w.z, w.w);
        }
    }
    const float bias = b_ih[j] + b_hh[j];
    const float wi0  = W_ih[2 * j + 0];
    const float wi1  = W_ih[2 * j + 1];

    if (j < H) h_lds[j] = 0.0f;
    float c = 0.0f;

    // Register double-buffered x staging: preload chunk 0.
    float xreg = (j < 2 * T) ? x[j] : 0.0f;   // 2*CHUNK == blockDim.x
    __syncthreads();

    const int nchunks = (T + CHUNK - 1) / CHUNK;
    for (int ch = 0; ch < nchunks; ++ch) {
        xbuf[j] = xreg;
        // Issue next chunk's global load now; 128 steps of work hide it.
        const int nidx = (ch + 1) * (2 * CHUNK) + j;
        if (ch + 1 < nchunks && nidx < 2 * T) xreg = x[nidx];
        __syncthreads();

        const int tend = (T - ch * CHUNK < CHUNK) ? (T - ch * CHUNK) : CHUNK;
        for (int tt = 0; tt < tend; ++tt) {
            const int t = ch * CHUNK + tt;
            const float x0 = xbuf[2 * tt + 0];
            const float x1 = xbuf[2 * tt + 1];

            v2f acc2 = mkv2(bias + x0 * wi0 + x1 * wi1, 0.0f);
            const float4* h4 = (const float4*)h_lds;
            #pragma unroll
            for (int k = 0; k < H / 4; ++k) {
                float4 hv = h4[k];                      // ds_load_b128, broadcast
                v2f hlo = mkv2(hv.x, hv.y);
                v2f hhi = mkv2(hv.z, hv.w);
                asm("v_pk_fma_f32 %0, %1, %2, %0" : "+v"(acc2) : "v"(hlo), "v"(w2[2 * k + 0]));
                asm("v_pk_fma_f32 %0, %1, %2, %0" : "+v"(acc2) : "v"(hhi), "v"(w2[2 * k + 1]));
            }
            gates_lds[j] = acc2.x + acc2.y;
            __syncthreads();

            if (j < H) {
                const float gi = fsigmoid(gates_lds[j]);
                const float gf = fsigmoid(gates_lds[H + j]);
                const float gg = ftanh(gates_lds[2 * H + j]);
                const float go = fsigmoid(gates_lds[3 * H + j]);
                c = gf * c + gi * gg;
                const float hn = go * ftanh(c);
                h_lds[j] = hn;
                hs[(long)t * H + j] = hn;               // fire-and-forget store
            }
            __syncthreads();
        }
    }
}

// ---------------------------------------------------------------------------
// Kernel B: out = hs @ fc_w^T + fc_b via V_WMMA_F32_16X16X4_F32.
// One wave per 16-row tile of hs; 16 WMMAs accumulate K=64; N padded 2->16.
// Tile index is wave-uniform so EXEC is all-ones at every WMMA.
// ---------------------------------------------------------------------------
__global__ __launch_bounds__(256, 4)
void fc_wmma_kernel(const float* __restrict__ hs,
                    const float* __restrict__ fc_w,
                    const float* __restrict__ fc_b,
                    float* __restrict__ out,
                    int T) {
    const int lane = threadIdx.x & 31;
    const int wave = threadIdx.x >> 5;
    const int tile = blockIdx.x * 8 + wave;
    const int row0 = tile * 16;
    if (row0 >= T) return;                   // wave-uniform exit

    const int M    = lane & 15;              // A-fragment row for this lane
    const int half = lane >> 4;              // 0: K={0,1}; 1: K={2,3}
    const int N    = lane & 15;              // B/D column for this lane

    // B fragment per K-block: B[k][n] = fc_w[n][k] (n<2), zero-padded to N=16.
    v2f bfrag[16];
    #pragma unroll
    for (int kk = 0; kk < 16; ++kk) {
        const int k0 = kk * 4 + half * 2;
        const float b0 = (N < 2) ? fc_w[N * H + k0 + 0] : 0.0f;
        const float b1 = (N < 2) ? fc_w[N * H + k0 + 1] : 0.0f;
        bfrag[kk] = mkv2(b0, b1);
    }

    int row = row0 + M;
    if (row >= T) row = T - 1;               // harmless clamp (T%16==0 normally)
    const float* arow = hs + (long)row * H;

    v8f cacc = {};
    #pragma unroll
    for (int kk = 0; kk < 16; ++kk) {
        const int k0 = kk * 4 + half * 2;
        const v2f afrag = *(const v2f*)(arow + k0);
        cacc = __builtin_amdgcn_wmma_f32_16x16x4_f32(
            /*neg_a=*/false, afrag, /*neg_b=*/false, bfrag[kk],
            /*c_mod=*/(short)0, cacc, /*reuse_a=*/false, /*reuse_b=*/false);
    }

    // D layout: VGPR r, lane -> M = r + 8*half, N = lane&15.
    #pragma unroll
    for (int r = 0; r < 8; ++r) {
        const int orow = row0 + r + 8 * half;
        if (N < 2 && orow < T) out[orow * 2 + N] = cacc[r] + fc_b[N];
    }
}

// ---------------------------------------------------------------------------
extern "C" void kernel_launch(void* const* d_in, const int* in_sizes, int n_in,
                              void* d_out, int out_size, void* d_ws, size_t ws_size,
                              hipStream_t stream) {
    const float* x    = (const float*)d_in[0];
    const float* W_ih = (const float*)d_in[1];
    const float* W_hh = (const float*)d_in[2];
    const float* b_ih = (const float*)d_in[3];
    const float* b_hh = (const float*)d_in[4];
    const float* fc_w = (const float*)d_in[5];
    const float* fc_b = (const float*)d_in[6];
    float* out = (float*)d_out;

    const int T = in_sizes[0] / 2;           // x is [1, T, 2]
    float* hs = (float*)d_ws;                // [T, 64] f32 scratch

    lstm_scan_kernel<<<1, 256, 0, stream>>>(x, W_ih, W_hh, b_ih, b_hh, hs, T);

    const int tiles  = (T + 15) / 16;
    const int blocks = (tiles + 7) / 8;
    fc_wmma_kernel<<<blocks, 256, 0, stream>>>(hs, fc_w, fc_b, out, T);
}